// correlation_loss2_58935541236205
// MI455X (gfx1250) — compile-verified
//
#include <hip/hip_runtime.h>
#include <hip/hip_bf16.h>
#include <math.h>

typedef float v2f __attribute__((ext_vector_type(2)));
typedef float v8f __attribute__((ext_vector_type(8)));
typedef int   v4i __attribute__((ext_vector_type(4)));

#define N_ROWS 8192
#define D_DIM  1024
#define T_TEMP 4.0f
#define INV_T  0.25f
#define A_OFF  0.025f   // 0.1 / T

#define BR 64           // rows per block
#define BC 128          // cols per column tile (2 col-groups of 64 per wave)
#define KC 32           // K chunk staged through LDS (double buffered)
#define LDA (KC + 4)    // padded LDS stride -> conflict-free b64 fragment loads
#define NCHUNK (D_DIM / KC)

// ---- async global->LDS staging (CDNA5) with safe fallback -------------------
typedef __attribute__((address_space(1))) v4i as1_v4i;   // global int4
typedef __attribute__((address_space(3))) v4i as3_v4i;   // LDS int4

#if defined(__gfx1250__) && __has_builtin(__builtin_amdgcn_global_load_async_to_lds_b128)
#define USE_ASYNC 1
#else
#define USE_ASYNC 0
#endif

__device__ __forceinline__ void copy16(const float* g, float* l) {
#if USE_ASYNC
    // launder through integers: global flat addr == global addr; flat LDS
    // pointer keeps the LDS byte offset in addr[31:0] (ISA 10.2).
    as1_v4i* gp = (as1_v4i*)(unsigned long long)(const void*)g;
    as3_v4i* lp = (as3_v4i*)(unsigned int)(unsigned long long)(const void*)l;
    __builtin_amdgcn_global_load_async_to_lds_b128(gp, lp, 0, 0);
#else
    *(float4*)l = *(const float4*)g;
#endif
}

#if USE_ASYNC
#if __has_builtin(__builtin_amdgcn_s_wait_asynccnt)
#define WAIT_ASYNC(n) __builtin_amdgcn_s_wait_asynccnt(n)
#else
#define WAIT_ASYNC(n) asm volatile("s_wait_asynccnt %0" ::"n"(n) : "memory")
#endif
#else
#define WAIT_ASYNC(n)
#endif
// -----------------------------------------------------------------------------

__global__ __launch_bounds__(256) void zero_counts_kernel(int* counts) {
    int i = blockIdx.x * blockDim.x + threadIdx.x;
    if (i < 1024) counts[i] = 0;
}

__global__ __launch_bounds__(256) void hist_kernel(const int* __restrict__ label,
                                                   int* __restrict__ counts) {
    int i = blockIdx.x * blockDim.x + threadIdx.x;
    if (i < N_ROWS) atomicAdd(&counts[label[i]], 1);
}

// Fused feat@featT + online row-softmax statistics + per-row loss.
__global__ __launch_bounds__(256) void gram_kl_kernel(const float* __restrict__ feat,
                                                      const int* __restrict__ label,
                                                      const int* __restrict__ counts,
                                                      float* __restrict__ partial) {
    __shared__ float As[2][BR * LDA];   // 64 x 36 x2
    __shared__ float Bs[2][BC * LDA];   // 128 x 36 x2
    __shared__ float stats[2][BR][4];
    __shared__ float wsum[2];

    const int tid  = threadIdx.x;
    const int wave = tid >> 5;
    const int lane = tid & 31;
    const int half = lane >> 4;        // which 16-lane half
    const int ln   = lane & 15;
    const int rowg = wave >> 1;        // 4 row groups of 16
    const int colg = wave & 1;         // 2 column groups of 64
    const int r0   = blockIdx.x * BR;

    // labels of the 8 rows this lane's accumulator slices cover
    int labRow[8];
#pragma unroll
    for (int v = 0; v < 8; ++v)
        labRow[v] = label[r0 + rowg * 16 + half * 8 + v];

    // online softmax stats per owned row
    float m[8], Z[8], Y[8], Zp[8];
#pragma unroll
    for (int v = 0; v < 8; ++v) { m[v] = -1.0e30f; Z[v] = 0.f; Y[v] = 0.f; Zp[v] = 0.f; }

    for (int c0 = 0; c0 < N_ROWS; c0 += BC) {
        v8f acc0 = {0.f,0.f,0.f,0.f,0.f,0.f,0.f,0.f};
        v8f acc1 = acc0, acc2 = acc0, acc3 = acc0;

        // ---- prologue: stage chunk 0 into buffer 0 (6 x b128 per thread) ----
        {
            const int kc = 0;
#pragma unroll
            for (int t = 0; t < 2; ++t) {            // A: 64 x 32 = 512 f4
                int idx = tid + t * 256;
                int r = idx >> 3, q = idx & 7;
                copy16(&feat[(size_t)(r0 + r) * D_DIM + kc + q * 4],
                       &As[0][r * LDA + q * 4]);
            }
#pragma unroll
            for (int t = 0; t < 4; ++t) {            // B: 128 x 32 = 1024 f4
                int idx = tid + t * 256;
                int r = idx >> 3, q = idx & 7;
                copy16(&feat[(size_t)(c0 + r) * D_DIM + kc + q * 4],
                       &Bs[0][r * LDA + q * 4]);
            }
        }

        int buf = 0;
        for (int kci = 0; kci < NCHUNK; ++kci) {
            const bool have_next = (kci + 1) < NCHUNK;
            __syncthreads();   // everyone finished reading buf^1 last iteration
            if (have_next) {
                const int kc = (kci + 1) * KC;
#pragma unroll
                for (int t = 0; t < 2; ++t) {
                    int idx = tid + t * 256;
                    int r = idx >> 3, q = idx & 7;
                    copy16(&feat[(size_t)(r0 + r) * D_DIM + kc + q * 4],
                           &As[buf ^ 1][r * LDA + q * 4]);
                }
#pragma unroll
                for (int t = 0; t < 4; ++t) {
                    int idx = tid + t * 256;
                    int r = idx >> 3, q = idx & 7;
                    copy16(&feat[(size_t)(c0 + r) * D_DIM + kc + q * 4],
                           &Bs[buf ^ 1][r * LDA + q * 4]);
                }
                WAIT_ASYNC(6);   // next chunk (6 ops) in flight; current drained
            } else {
                WAIT_ASYNC(0);
            }
            __syncthreads();   // chunk kci visible to all waves

            // ISA f32 A-layout (16x4): lanes 0-15 -> K=kk,kk+1 ; 16-31 -> kk+2,kk+3
            const float* Ab = &As[buf][(rowg * 16 + ln) * LDA + half * 2];
            const float* Bb = &Bs[buf][(colg * 64 + ln) * LDA + half * 2];
#pragma unroll
            for (int kk = 0; kk < KC; kk += 4) {
                v2f a  = *(const v2f*)(Ab + kk);
                v2f b0 = *(const v2f*)(Bb + kk);
                v2f b1 = *(const v2f*)(Bb + 16 * LDA + kk);
                v2f b2 = *(const v2f*)(Bb + 32 * LDA + kk);
                v2f b3 = *(const v2f*)(Bb + 48 * LDA + kk);
                acc0 = __builtin_amdgcn_wmma_f32_16x16x4_f32(false, a, false, b0, (short)0, acc0, false, false);
                acc1 = __builtin_amdgcn_wmma_f32_16x16x4_f32(false, a, false, b1, (short)0, acc1, false, false);
                acc2 = __builtin_amdgcn_wmma_f32_16x16x4_f32(false, a, false, b2, (short)0, acc2, false, false);
                acc3 = __builtin_amdgcn_wmma_f32_16x16x4_f32(false, a, false, b3, (short)0, acc3, false, false);
            }
            buf ^= 1;
        }

        // online update with this 64-column strip (4 slabs of 16 columns)
#pragma unroll
        for (int t = 0; t < 4; ++t) {
            int labCol = label[c0 + colg * 64 + t * 16 + ln];
            const v8f& acc = (t == 0) ? acc0 : (t == 1) ? acc1 : (t == 2) ? acc2 : acc3;
#pragma unroll
            for (int v = 0; v < 8; ++v) {
                float s  = acc[v];
                float m2 = fmaxf(m[v], s);
                float r  = __expf((m[v] - m2) * INV_T);
                float e  = __expf((s    - m2) * INV_T);
                Z[v]  = Z[v]  * r + e;
                Y[v]  = Y[v]  * r + e * (s * INV_T);
                Zp[v] = Zp[v] * r + ((labCol == labRow[v]) ? e : 0.f);
                m[v]  = m2;
            }
        }
    }

    // merge stats across the 16 column lanes (butterfly stays inside each half)
#pragma unroll
    for (int mask = 1; mask <= 8; mask <<= 1) {
#pragma unroll
        for (int v = 0; v < 8; ++v) {
            float om  = __shfl_xor(m[v],  mask, 32);
            float oZ  = __shfl_xor(Z[v],  mask, 32);
            float oY  = __shfl_xor(Y[v],  mask, 32);
            float oZp = __shfl_xor(Zp[v], mask, 32);
            float m2 = fmaxf(m[v], om);
            float ra = __expf((m[v] - m2) * INV_T);
            float rb = __expf((om   - m2) * INV_T);
            Z[v]  = Z[v]  * ra + oZ  * rb;
            Y[v]  = Y[v]  * ra + oY  * rb;
            Zp[v] = Zp[v] * ra + oZp * rb;
            m[v]  = m2;
        }
    }
    if (ln == 0) {
#pragma unroll
        for (int v = 0; v < 8; ++v) {
            int row = rowg * 16 + half * 8 + v;
            stats[colg][row][0] = m[v];
            stats[colg][row][1] = Z[v];
            stats[colg][row][2] = Y[v];
            stats[colg][row][3] = Zp[v];
        }
    }
    __syncthreads();

    // finalize: one thread per row (64 rows -> waves 0,1), merge col groups
    float loss = 0.f;
    if (tid < BR) {
        float mm = stats[0][tid][0], ZZ = stats[0][tid][1];
        float YY = stats[0][tid][2], PP = stats[0][tid][3];
        {
            float om = stats[1][tid][0], oZ = stats[1][tid][1];
            float oY = stats[1][tid][2], oP = stats[1][tid][3];
            float m2 = fmaxf(mm, om);
            float ra = __expf((mm - m2) * INV_T);
            float rb = __expf((om - m2) * INV_T);
            ZZ = ZZ * ra + oZ * rb;
            YY = YY * ra + oY * rb;
            PP = PP * ra + oP * rb;
            mm = m2;
        }
        int   gr = r0 + tid;
        int   k  = counts[label[gr]];
        float Zs = (float)k * __expf(-A_OFF) + (float)(N_ROWS - k) * __expf(A_OFF);
        loss = YY / ZZ - mm * INV_T - __logf(ZZ)
             - A_OFF + 2.0f * A_OFF * (PP / ZZ) + __logf(Zs);
#pragma unroll
        for (int mask = 1; mask <= 16; mask <<= 1)
            loss += __shfl_xor(loss, mask, 32);
        if (lane == 0) wsum[wave] = loss;
    }
    __syncthreads();
    if (tid == 0) partial[blockIdx.x] = wsum[0] + wsum[1];
}

__global__ __launch_bounds__(128) void final_kernel(const float* __restrict__ partial,
                                                    float* __restrict__ out) {
    __shared__ float buf[128];
    buf[threadIdx.x] = partial[threadIdx.x];
    __syncthreads();
    for (int s = 64; s > 0; s >>= 1) {
        if (threadIdx.x < s) buf[threadIdx.x] += buf[threadIdx.x + s];
        __syncthreads();
    }
    if (threadIdx.x == 0)
        out[0] = buf[0] * (T_TEMP * T_TEMP / (float)N_ROWS);
}

extern "C" void kernel_launch(void* const* d_in, const int* in_sizes, int n_in,
                              void* d_out, int out_size, void* d_ws, size_t ws_size,
                              hipStream_t stream) {
    const float* feat  = (const float*)d_in[0];
    const int*   label = (const int*)d_in[1];

    int*   counts  = (int*)d_ws;               // 1024 ints (1000 classes, padded)
    float* partial = (float*)d_ws + 1024;      // 128 block partials

    zero_counts_kernel<<<4, 256, 0, stream>>>(counts);
    hist_kernel<<<N_ROWS / 256, 256, 0, stream>>>(label, counts);
    gram_kl_kernel<<<N_ROWS / BR, 256, 0, stream>>>(feat, label, counts, partial);
    final_kernel<<<1, 128, 0, stream>>>(partial, (float*)d_out);
}